// GNNEncoder_28853590295049
// MI455X (gfx1250) — compile-verified
//
#include <hip/hip_runtime.h>
#include <hip/hip_bf16.h>

#define CH 128
#define BN_EPS 1e-5f
#define LDSPAD 132   // 16-row A tile stride; 132 % 64 = 4 -> conflict-free column reads

typedef __attribute__((ext_vector_type(2))) float v2f;
typedef __attribute__((ext_vector_type(8))) float v8f;

// ---------------- degree / norm ----------------
__global__ void k_deg_init(float* __restrict__ deg, int n) {
    int i = blockIdx.x * blockDim.x + threadIdx.x;
    if (i < n) deg[i] = 1.0f;               // self loop contributes 1
}

__global__ void k_deg_accum(const int* __restrict__ ei, float* __restrict__ deg, int E) {
    int e = blockIdx.x * blockDim.x + threadIdx.x;
    if (e < E) atomicAdd(&deg[ei[E + e]], 1.0f);   // dst row
}

__global__ void k_dinv(float* __restrict__ deg, int n) {
    int i = blockIdx.x * blockDim.x + threadIdx.x;
    if (i < n) deg[i] = rsqrtf(deg[i]);     // deg >= 1 always
}

// ---------------- layer 0: x(N,5) @ W0(5,128), fused aggregation seed ----------------
// hOut = x@W0 ; hAgg = bias + hOut * dinv^2  (self-loop term, written non-atomically)
__global__ void __launch_bounds__(256) k_lin0(const float* __restrict__ x,
                                              const float* __restrict__ W0,
                                              const float* __restrict__ bias,
                                              const float* __restrict__ dinv,
                                              float* __restrict__ hOut,
                                              float* __restrict__ hAgg, int n) {
    __shared__ float sW[5 * CH];
    for (int i = threadIdx.x; i < 5 * CH; i += blockDim.x) sW[i] = W0[i];
    __syncthreads();
    int idx = blockIdx.x * blockDim.x + threadIdx.x;
    if (idx >= n * CH) return;
    int node = idx >> 7, c = idx & (CH - 1);
    const float* xr = x + node * 5;
    float acc = 0.0f;
#pragma unroll
    for (int k = 0; k < 5; ++k) acc = fmaf(xr[k], sW[k * CH + c], acc);
    float d = dinv[node];
    hOut[idx] = acc;
    hAgg[idx] = fmaf(acc, d * d, bias[c]);
}

// ---------------- WMMA GEMM: A(N,128) @ W(128,128), fused aggregation seed -------------
// Block = 256 threads = 8 waves. Block owns 16 rows; wave w owns cols [16w,16w+16).
// Writes hOut = A@W and hAgg = bias + hOut*dinv^2. Safe to alias hAgg with A:
// each block fully stages its A rows in LDS before any store, and only writes rows it owns.
__global__ void __launch_bounds__(256, 2) k_gemm_wmma(const float* __restrict__ A,
                                                      const float* __restrict__ W,
                                                      const float* __restrict__ bias,
                                                      const float* __restrict__ dinv,
                                                      float* __restrict__ hOut,
                                                      float* __restrict__ hAgg, int n) {
    __shared__ float As[16 * LDSPAD];
    int m0 = blockIdx.x * 16;
    for (int i = threadIdx.x; i < 16 * CH; i += blockDim.x) {
        int r = i >> 7, c = i & (CH - 1);
        int row = m0 + r;
        As[r * LDSPAD + c] = (row < n) ? A[(size_t)row * CH + c] : 0.0f;
    }
    __syncthreads();

    int wave = threadIdx.x >> 5;
    int lane = threadIdx.x & 31;
    int n0   = wave << 4;
    int mlo  = lane & 15;            // M (for A) / N (for B,D)
    int khi  = (lane >> 4) << 1;     // hi half-wave holds K+2,K+3

    v8f acc = {};
#pragma unroll
    for (int k = 0; k < CH; k += 4) {
        v2f a, b;
        a.x = As[mlo * LDSPAD + k + khi];
        a.y = As[mlo * LDSPAD + k + khi + 1];
        b.x = W[(k + khi) * CH + n0 + mlo];
        b.y = W[(k + khi + 1) * CH + n0 + mlo];
        // V_WMMA_F32_16X16X4_F32 : D = A(16x4) x B(4x16) + C
        acc = __builtin_amdgcn_wmma_f32_16x16x4_f32(
            /*neg_a=*/false, a, /*neg_b=*/false, b,
            /*c_mod=*/(short)0, acc, /*reuse_a=*/false, /*reuse_b=*/false);
    }

    float bc = bias[n0 + mlo];
    int rbase = (lane >> 4) << 3;    // lanes 16-31 hold rows M = r+8
#pragma unroll
    for (int r = 0; r < 8; ++r) {
        int row = m0 + rbase + r;
        if (row < n) {
            float d = dinv[row];
            size_t o = (size_t)row * CH + n0 + mlo;
            hOut[o] = acc[r];
            hAgg[o] = fmaf(acc[r], d * d, bc);
        }
    }
}

// ---------------- edge scatter ----------------
// one wave32 per edge: lane holds 4 channels (float4 gather, 4 f32 atomics scatter)
__global__ void __launch_bounds__(256) k_msg_edges(const int* __restrict__ ei,
                                                   const float* __restrict__ dinv,
                                                   const float* __restrict__ h,
                                                   float* __restrict__ out, int E) {
    int e = blockIdx.x * (blockDim.x >> 5) + (threadIdx.x >> 5);
    int lane = threadIdx.x & 31;
    if (e >= E) return;
    int src = ei[e];
    int dst = ei[E + e];
    float norm = dinv[src] * dinv[dst];
    const float4* hs = (const float4*)(h + (size_t)src * CH);
    float4 v = hs[lane];
    float* o = out + (size_t)dst * CH + lane * 4;
    atomicAdd(o + 0, v.x * norm);
    atomicAdd(o + 1, v.y * norm);
    atomicAdd(o + 2, v.z * norm);
    atomicAdd(o + 3, v.w * norm);
}

// ---------------- batchnorm ----------------
__global__ void k_zero(float* __restrict__ p, int n) {
    for (int i = blockIdx.x * blockDim.x + threadIdx.x; i < n; i += gridDim.x * blockDim.x)
        p[i] = 0.0f;
}

__global__ void __launch_bounds__(256) k_bn_stats(const float* __restrict__ h,
                                                  float* __restrict__ stats, int n) {
    __shared__ float s[2 * CH];
    for (int i = threadIdx.x; i < 2 * CH; i += blockDim.x) s[i] = 0.0f;
    __syncthreads();
    int total = n * CH;
    for (int idx = blockIdx.x * blockDim.x + threadIdx.x; idx < total;
         idx += gridDim.x * blockDim.x) {
        float v = h[idx];
        int c = idx & (CH - 1);
        atomicAdd(&s[c], v);            // ds_add_f32
        atomicAdd(&s[CH + c], v * v);
    }
    __syncthreads();
    for (int i = threadIdx.x; i < 2 * CH; i += blockDim.x) atomicAdd(&stats[i], s[i]);
}

__global__ void __launch_bounds__(256) k_bn_apply(float* __restrict__ h,
                                                  const float* __restrict__ stats,
                                                  const float* __restrict__ gamma,
                                                  const float* __restrict__ beta, int n) {
    int idx = blockIdx.x * blockDim.x + threadIdx.x;
    if (idx >= n * CH) return;
    int c = idx & (CH - 1);
    float invn = 1.0f / (float)n;
    float mean = stats[c] * invn;
    float var  = stats[CH + c] * invn - mean * mean;
    float v = (h[idx] - mean) * rsqrtf(var + BN_EPS) * gamma[c] + beta[c];
    h[idx] = fmaxf(v, 0.0f);
}

// ---------------- pooling + head ----------------
__global__ void __launch_bounds__(256) k_pool(const float* __restrict__ h,
                                              const int* __restrict__ batch,
                                              float* __restrict__ pool,
                                              float* __restrict__ cnt, int n) {
    int idx = blockIdx.x * blockDim.x + threadIdx.x;
    if (idx >= n * CH) return;
    int node = idx >> 7, c = idx & (CH - 1);
    int b = batch[node];
    atomicAdd(&pool[b * CH + c], h[idx]);
    if (c == 0) atomicAdd(&cnt[b], 1.0f);
}

__global__ void __launch_bounds__(64) k_head(const float* __restrict__ pool,
                                             const float* __restrict__ cnt,
                                             const float* __restrict__ emb_W,
                                             const float* __restrict__ emb_b,
                                             float* __restrict__ out, int ngraphs) {
    int g = blockIdx.x, j = threadIdx.x;
    if (g >= ngraphs) return;
    float ic = 1.0f / fmaxf(cnt[g], 1.0f);
    float acc = 0.0f;
#pragma unroll 4
    for (int c = 0; c < CH; ++c)
        acc = fmaf(pool[g * CH + c] * ic, emb_W[c * 64 + j], acc);
    out[g * 64 + j] = acc + emb_b[j];
}

// ---------------- launcher ----------------
extern "C" void kernel_launch(void* const* d_in, const int* in_sizes, int n_in,
                              void* d_out, int out_size, void* d_ws, size_t ws_size,
                              hipStream_t stream) {
    const float* x     = (const float*)d_in[0];
    const float* W0    = (const float*)d_in[1];
    const float* Wh    = (const float*)d_in[2];
    const float* bias  = (const float*)d_in[3];
    const float* gamma = (const float*)d_in[4];
    const float* beta  = (const float*)d_in[5];
    const float* emb_W = (const float*)d_in[6];
    const float* emb_b = (const float*)d_in[7];
    const int*   ei    = (const int*)d_in[8];
    const int*   batch = (const int*)d_in[9];

    int n       = in_sizes[9];       // N_NODES
    int E       = in_sizes[8] / 2;   // N_EDGES
    int ngraphs = out_size / 64;

    // workspace layout
    size_t npad  = ((size_t)n + 255) & ~(size_t)255;
    float* dinv  = (float*)d_ws;
    float* hA    = dinv + npad;                 // h after matmul (gather source)
    float* hB    = hA + npad * CH;              // aggregation target / post-BN h
    float* stats = hB + npad * CH;
    float* pool  = stats + 2 * CH;
    float* cnt   = pool + (size_t)ngraphs * CH;
    (void)ws_size; (void)n_in;

    int gN    = (n + 255) / 256;
    int gE    = (E + 255) / 256;
    int gElem = (int)(((size_t)n * CH + 255) / 256);
    int gEdge = (E + 7) / 8;                 // 8 edges per 256-thread block
    int mtiles = (n + 15) / 16;

    // degree -> dinv (stored in-place)
    k_deg_init <<<gN, 256, 0, stream>>>(dinv, n);
    k_deg_accum<<<gE, 256, 0, stream>>>(ei, dinv, E);
    k_dinv     <<<gN, 256, 0, stream>>>(dinv, n);

    for (int l = 0; l < 3; ++l) {
        if (l == 0) {
            // hA = x @ W0 ; hB = bias0 + hA*dinv^2
            k_lin0<<<gElem, 256, 0, stream>>>(x, W0, bias, dinv, hA, hB, n);
        } else {
            // hA = hB @ Wh[l-1] ; hB = bias_l + hA*dinv^2   (f32 WMMA, in-place safe)
            k_gemm_wmma<<<mtiles, 256, 0, stream>>>(hB, Wh + (size_t)(l - 1) * CH * CH,
                                                    bias + l * CH, dinv, hA, hB, n);
        }
        // edge scatter: hB += norm * hA[src]
        k_msg_edges<<<gEdge, 256, 0, stream>>>(ei, dinv, hA, hB, E);
        // batchnorm + relu (in place on hB)
        k_zero    <<<1, 256, 0, stream>>>(stats, 2 * CH);
        k_bn_stats<<<1024, 256, 0, stream>>>(hB, stats, n);
        k_bn_apply<<<gElem, 256, 0, stream>>>(hB, stats, gamma + l * CH, beta + l * CH, n);
    }

    // global mean pool + head
    k_zero<<<(ngraphs * (CH + 1) + 255) / 256, 256, 0, stream>>>(pool, ngraphs * (CH + 1));
    k_pool<<<gElem, 256, 0, stream>>>(hB, batch, pool, cnt, n);
    k_head<<<ngraphs, 64, 0, stream>>>(pool, cnt, emb_W, emb_b, (float*)d_out, ngraphs);
}